// KNN_model_23991687315673
// MI455X (gfx1250) — compile-verified
//
#include <hip/hip_runtime.h>
#include <hip/hip_bf16.h>

typedef __attribute__((ext_vector_type(2))) float v2f;
typedef __attribute__((ext_vector_type(8))) float v8f;

#define HW 65536
#define IMG 256

// ---------------------------------------------------------------------------
// Conv 3x3 SAME + bias. Block: 256 threads -> 16x16 spatial tile x 16 out chans.
// Input channels staged through LDS in chunks of <=16.
// ---------------------------------------------------------------------------
template <int CIN>
__global__ __launch_bounds__(256) void conv3x3_kernel(
    const float* __restrict__ in,    // [CIN,256,256]
    const float* __restrict__ w,     // [COUT,CIN,3,3]
    const float* __restrict__ bias,  // [COUT]
    int cout,
    float* __restrict__ out) {       // [COUT,256,256]
  constexpr int CC = (CIN < 16) ? CIN : 16;
  __shared__ float lin[CC][18][18];
  __shared__ float lw[16][CC][9];

  const int tx = threadIdx.x & 15;
  const int ty = threadIdx.x >> 4;
  const int gx0 = blockIdx.x * 16;
  const int gy0 = blockIdx.y * 16;
  const int oc0 = blockIdx.z * 16;
  const int noc = (cout - oc0 < 16) ? (cout - oc0) : 16;

  float acc[16];
#pragma unroll
  for (int i = 0; i < 16; ++i) acc[i] = 0.0f;

  for (int c0 = 0; c0 < CIN; c0 += CC) {
    // cooperative load of (CC x 18 x 18) input tile with zero padding
    for (int idx = threadIdx.x; idx < CC * 324; idx += 256) {
      int c = idx / 324, rem = idx % 324;
      int yy = rem / 18, xx = rem % 18;
      int gy = gy0 + yy - 1, gx = gx0 + xx - 1;
      float v = 0.0f;
      if (gy >= 0 && gy < IMG && gx >= 0 && gx < IMG)
        v = in[(size_t)(c0 + c) * HW + gy * IMG + gx];
      lin[c][yy][xx] = v;
    }
    // cooperative load of weights for this (oc-chunk, cin-chunk)
    for (int idx = threadIdx.x; idx < noc * CC * 9; idx += 256) {
      int oc = idx / (CC * 9), rem = idx % (CC * 9);
      int c = rem / 9, t = rem % 9;
      lw[oc][c][t] = w[((size_t)(oc0 + oc) * CIN + (c0 + c)) * 9 + t];
    }
    __syncthreads();

#pragma unroll 1
    for (int c = 0; c < CC; ++c) {
#pragma unroll
      for (int t = 0; t < 9; ++t) {
        const int dy = t / 3, dx = t % 3;
        const float v = lin[c][ty + dy][tx + dx];
#pragma unroll
        for (int oc = 0; oc < 16; ++oc) acc[oc] += v * lw[oc][c][t];
      }
    }
    __syncthreads();
  }

  const int gy = gy0 + ty, gx = gx0 + tx;
  for (int oc = 0; oc < noc; ++oc)
    out[(size_t)(oc0 + oc) * HW + gy * IMG + gx] = acc[oc] + bias[oc0 + oc];
}

// ---------------------------------------------------------------------------
// BN stage 1: per-(channel, block) partial sum / sumsq. 64 blocks per channel,
// each reduces 1024 elements. Deterministic tree reduction.
// ---------------------------------------------------------------------------
__global__ __launch_bounds__(256) void bn_stats_kernel(
    const float* __restrict__ y, float* __restrict__ psum,
    float* __restrict__ psumsq) {
  const int c = blockIdx.y;
  const int blk = blockIdx.x;  // 0..63
  const float* p = y + (size_t)c * HW + blk * 1024;
  float s = 0.0f, ss = 0.0f;
  for (int i = threadIdx.x; i < 1024; i += 256) {
    float v = p[i];
    s += v;
    ss += v * v;
  }
  __shared__ float ls[256], lss[256];
  ls[threadIdx.x] = s;
  lss[threadIdx.x] = ss;
  __syncthreads();
  for (int off = 128; off > 0; off >>= 1) {
    if (threadIdx.x < off) {
      ls[threadIdx.x] += ls[threadIdx.x + off];
      lss[threadIdx.x] += lss[threadIdx.x + off];
    }
    __syncthreads();
  }
  if (threadIdx.x == 0) {
    psum[c * 64 + blk] = ls[0];
    psumsq[c * 64 + blk] = lss[0];
  }
}

// BN stage 2: fold partials -> scale/shift per channel
__global__ void bn_finalize_kernel(const float* __restrict__ psum,
                                   const float* __restrict__ psumsq,
                                   const float* __restrict__ gamma,
                                   const float* __restrict__ beta,
                                   float* __restrict__ scale,
                                   float* __restrict__ shift) {
  const int c = blockIdx.x;
  if (threadIdx.x == 0) {
    float s = 0.0f, ss = 0.0f;
    for (int i = 0; i < 64; ++i) {
      s += psum[c * 64 + i];
      ss += psumsq[c * 64 + i];
    }
    const float inv_n = 1.0f / 65536.0f;
    const float mean = s * inv_n;
    const float var = ss * inv_n - mean * mean;  // biased, like torch BN fwd
    const float sc = gamma[c] * rsqrtf(var + 1e-5f);
    scale[c] = sc;
    shift[c] = beta[c] - mean * sc;
  }
}

// BN apply + ReLU, in place
__global__ __launch_bounds__(256) void bn_relu_kernel(
    float* __restrict__ y, const float* __restrict__ scale,
    const float* __restrict__ shift) {
  const int c = blockIdx.y;
  const size_t i = (size_t)c * HW + blockIdx.x * 256 + threadIdx.x;
  float v = y[i] * scale[c] + shift[c];
  y[i] = fmaxf(v, 0.0f);
}

// ---------------------------------------------------------------------------
// Patch extraction: feat [2,256,256] -> P[8192][16] row-major + sq[8192]
// n = (c, hb, wb); P[n][r*4+col] = feat[c][hb*4+r][wb*4+col]
// ---------------------------------------------------------------------------
__global__ __launch_bounds__(256) void patch_kernel(
    const float* __restrict__ f, float* __restrict__ P,
    float* __restrict__ sq) {
  const int n = blockIdx.x * 256 + threadIdx.x;
  if (n >= 8192) return;
  const int c = n >> 12;
  const int hb = (n >> 6) & 63;
  const int wb = n & 63;
  const float* base = f + (size_t)c * HW + hb * 4 * IMG + wb * 4;
  float vals[16];
#pragma unroll
  for (int r = 0; r < 4; ++r) {
    const float4 v = *(const float4*)(base + r * IMG);
    vals[r * 4 + 0] = v.x;
    vals[r * 4 + 1] = v.y;
    vals[r * 4 + 2] = v.z;
    vals[r * 4 + 3] = v.w;
  }
  float s = 0.0f;
#pragma unroll
  for (int e = 0; e < 16; ++e) {
    P[(size_t)n * 16 + e] = vals[e];
    s += vals[e] * vals[e];
  }
  sq[n] = s;
}

// ---------------------------------------------------------------------------
// Pairwise distance via WMMA f32 16x16x4 chained over K=16.
// One wave32 computes one 16x16 output tile: D = sqrt(max(sqi + sqj - 2*P@P^T, 0))
// A fragment (16x4 f32): lanes 0-15 rows M=0..15 K={k0,k0+1}; lanes 16-31 K={k0+2,k0+3}
// B fragment (4x16 f32) of P^T has the identical per-lane load pattern with j-base.
// Epilogue: raw v_sqrt_f32 (<=1 ulp, inputs well-scaled) and NT stores so the
// 256MB one-shot output stream does not evict the L2 working set.
// ---------------------------------------------------------------------------
__global__ __launch_bounds__(256) void dist_kernel(
    const float* __restrict__ P, const float* __restrict__ sq,
    float* __restrict__ D) {
  const int lane = threadIdx.x & 31;
  const int gwave = blockIdx.x * (blockDim.x >> 5) + (threadIdx.x >> 5);
  const int ti = gwave >> 9;  // 512 tiles per dim
  const int tj = gwave & 511;
  const int i0 = ti << 4, j0 = tj << 4;
  const int r = lane & 15;
  const int half = lane >> 4;  // 0 or 1

  const float* arow = P + (size_t)(i0 + r) * 16 + half * 2;
  const float* brow = P + (size_t)(j0 + r) * 16 + half * 2;

  v8f c = {};
#pragma unroll
  for (int k0 = 0; k0 < 16; k0 += 4) {
    v2f a, b;
    a.x = arow[k0];
    a.y = arow[k0 + 1];
    b.x = brow[k0];
    b.y = brow[k0 + 1];
    // D = A(16x4) * B(4x16) + C  (fp32, RNE -- matches reference math)
    c = __builtin_amdgcn_wmma_f32_16x16x4_f32(false, a, false, b, (short)0, c,
                                              false, false);
  }

  const float sqj = sq[j0 + r];
  const float* sibase = sq + i0 + half * 8;
  float si[8];
#pragma unroll
  for (int rr = 0; rr < 8; ++rr) si[rr] = sibase[rr];

  float* drow = D + (size_t)(i0 + half * 8) * 8192 + j0 + r;
#pragma unroll
  for (int rr = 0; rr < 8; ++rr) {
    const float d2 = fmaxf(si[rr] + sqj - 2.0f * c[rr], 0.0f);
    // raw v_sqrt_f32; sqrt(0)==0 reproduces the reference's masked-sqrt
    const float dd = __builtin_amdgcn_sqrtf(d2);
    __builtin_nontemporal_store(dd, drow + (size_t)rr * 8192);
  }
}

// ---------------------------------------------------------------------------
// Host-side launch sequence
// ---------------------------------------------------------------------------
extern "C" void kernel_launch(void* const* d_in, const int* in_sizes, int n_in,
                              void* d_out, int out_size, void* d_ws,
                              size_t ws_size, hipStream_t stream) {
  (void)in_sizes; (void)n_in; (void)out_size; (void)ws_size;

  const float* x = (const float*)d_in[0];
  const float* W[4];
  const float* B[4];
  const float* G[4];
  const float* BE[4];
  for (int i = 0; i < 4; ++i) {
    W[i] = (const float*)d_in[1 + 4 * i];
    B[i] = (const float*)d_in[2 + 4 * i];
    G[i] = (const float*)d_in[3 + 4 * i];
    BE[i] = (const float*)d_in[4 + 4 * i];
  }
  float* out = (float*)d_out;

  // workspace carving (floats)
  float* ws = (float*)d_ws;
  float* bufA = ws;                      // 64*65536
  float* bufB = bufA + (size_t)64 * HW;  // 64*65536
  float* bufC = bufB + (size_t)64 * HW;  // 2*65536
  float* psum = bufC + (size_t)2 * HW;   // 64*64
  float* psumsq = psum + 64 * 64;        // 64*64
  float* scale = psumsq + 64 * 64;       // 64
  float* shift = scale + 64;             // 64
  float* Pm = shift + 64;                // 8192*16
  float* sqv = Pm + (size_t)8192 * 16;   // 8192

  dim3 blk(256);
  dim3 cgrid(16, 16, 4);  // 16x16 spatial tiles, 64/16 oc chunks

  // ---- layer 0: 3 -> 64 ----
  conv3x3_kernel<3><<<cgrid, blk, 0, stream>>>(x, W[0], B[0], 64, bufA);
  bn_stats_kernel<<<dim3(64, 64), blk, 0, stream>>>(bufA, psum, psumsq);
  bn_finalize_kernel<<<dim3(64), dim3(32), 0, stream>>>(psum, psumsq, G[0],
                                                        BE[0], scale, shift);
  bn_relu_kernel<<<dim3(256, 64), blk, 0, stream>>>(bufA, scale, shift);

  // ---- layer 1: 64 -> 64 ----
  conv3x3_kernel<64><<<cgrid, blk, 0, stream>>>(bufA, W[1], B[1], 64, bufB);
  bn_stats_kernel<<<dim3(64, 64), blk, 0, stream>>>(bufB, psum, psumsq);
  bn_finalize_kernel<<<dim3(64), dim3(32), 0, stream>>>(psum, psumsq, G[1],
                                                        BE[1], scale, shift);
  bn_relu_kernel<<<dim3(256, 64), blk, 0, stream>>>(bufB, scale, shift);

  // ---- layer 2: 64 -> 64 ----
  conv3x3_kernel<64><<<cgrid, blk, 0, stream>>>(bufB, W[2], B[2], 64, bufA);
  bn_stats_kernel<<<dim3(64, 64), blk, 0, stream>>>(bufA, psum, psumsq);
  bn_finalize_kernel<<<dim3(64), dim3(32), 0, stream>>>(psum, psumsq, G[2],
                                                        BE[2], scale, shift);
  bn_relu_kernel<<<dim3(256, 64), blk, 0, stream>>>(bufA, scale, shift);

  // ---- layer 3: 64 -> 2 ----
  conv3x3_kernel<64><<<dim3(16, 16, 1), blk, 0, stream>>>(bufA, W[3], B[3], 2,
                                                          bufC);
  bn_stats_kernel<<<dim3(64, 2), blk, 0, stream>>>(bufC, psum, psumsq);
  bn_finalize_kernel<<<dim3(2), dim3(32), 0, stream>>>(psum, psumsq, G[3],
                                                       BE[3], scale, shift);
  bn_relu_kernel<<<dim3(256, 2), blk, 0, stream>>>(bufC, scale, shift);

  // ---- patches + squared norms ----
  patch_kernel<<<dim3(32), blk, 0, stream>>>(bufC, Pm, sqv);

  // ---- pairwise distances: 512x512 tiles, 8 waves (tiles) per block ----
  dist_kernel<<<dim3(32768), blk, 0, stream>>>(Pm, sqv, out);
}